// GlobalPositionEncoding_19224273616920
// MI455X (gfx1250) — compile-verified
//
#include <hip/hip_runtime.h>
#include <hip/hip_bf16.h>

// Shapes from the reference
#define BB 16
#define NN 256
#define TT 96
#define HH 128
#define LN_EPS 1e-5f

typedef __attribute__((ext_vector_type(2))) float v2f;
typedef __attribute__((ext_vector_type(4))) float v4f;
typedef __attribute__((ext_vector_type(8))) float v8f;

// ---------------------------------------------------------------------------
// Generic f32 WMMA GEMM: C[M,N] = A[M,K] * B[K,N] (+ bias[N]) (+ add_rows[row % add_mod, N])
// One wave (32 lanes) computes one 16x16 tile of C using V_WMMA_F32_16X16X4_F32.
// M % 16 == 0, N % 16 == 0, K % 4 == 0. blockDim = 32, grid = (N/16, M/16).
//
// VGPR layouts per CDNA5 ISA 7.12.2:
//   A (16x4):  lane<16 holds row=lane, K=kk+0/+1 in reg0/reg1;
//              lane>=16 holds row=lane-16, K=kk+2/+3.
//   B (4x16):  reg0 = rows K=kk+0 (lanes 0-15) / K=kk+2 (lanes 16-31), col = lane&15
//              reg1 = rows K=kk+1 / K=kk+3.
//   C (16x16): reg r, lane l -> row r + 8*(l>=16), col l&15.
// NT_A: A is a one-time streaming read (use non-temporal loads).
// ---------------------------------------------------------------------------
template <bool NT_A>
__global__ void wmma_gemm_f32(const float* __restrict__ A,
                              const float* __restrict__ B,
                              float* __restrict__ C,
                              int M, int N, int K,
                              const float* __restrict__ bias,
                              const float* __restrict__ add_rows,
                              int add_mod) {
    const int tj   = blockIdx.x;          // N tile
    const int ti   = blockIdx.y;          // M tile
    const int lane = threadIdx.x & 31;
    const int half = lane >> 4;           // 0: lanes 0-15, 1: lanes 16-31
    const int lm   = lane & 15;

    const int arow = ti * 16 + lm;        // A row held by this lane
    const int bcol = tj * 16 + lm;        // B/C column held by this lane

    v8f acc = {};
    for (int kk = 0; kk < K; kk += 4) {
        const int ka = kk + (half << 1);
        v2f a, b;
        if (NT_A) {
            a[0] = __builtin_nontemporal_load(A + arow * K + ka);
            a[1] = __builtin_nontemporal_load(A + arow * K + ka + 1);
        } else {
            a[0] = A[arow * K + ka];
            a[1] = A[arow * K + ka + 1];
        }
        b[0] = B[ka * N + bcol];
        b[1] = B[(ka + 1) * N + bcol];
        acc = __builtin_amdgcn_wmma_f32_16x16x4_f32(
            /*neg_a=*/false, a, /*neg_b=*/false, b,
            /*c_mod=*/(short)0, acc, /*reuse_a=*/false, /*reuse_b=*/false);
    }

    const float cb = bias ? bias[bcol] : 0.0f;
#pragma unroll
    for (int r = 0; r < 8; ++r) {
        const int row = ti * 16 + r + (half << 3);
        float v = acc[r] + cb;
        if (add_rows) v += add_rows[(row % add_mod) * N + bcol];
        C[row * N + bcol] = v;
    }
}

// ---------------------------------------------------------------------------
// Fused broadcast + ReLU + LayerNorm + affine.
//   out[bn, t, :] = LN(relu(combined[bn,:] + timeW[t,:])) * gamma + beta
// One block per (b,n) pair (4096 blocks), 256 threads = 8 waves.
// Wave w handles t = w, w+8, ..., w+88 (12 rows). Lane owns 4 contiguous
// channels. Mean/var via wave32 xor-shuffle reduction.
//
// HBM traffic ~= 201 MB of 16B stores. The output stream (201 MB) exceeds the
// 192 MB L2, so stores use the NT temporal hint to avoid evicting the hot
// read set (combined 2 MB + timeW 48 KB + gamma/beta), which stays L2/WGP$
// resident across all 4096 blocks.
// ---------------------------------------------------------------------------
__global__ void fused_bias_relu_ln(const float* __restrict__ combined,
                                   const float* __restrict__ timeW,
                                   const float* __restrict__ gamma,
                                   const float* __restrict__ beta,
                                   float* __restrict__ out) {
    const int bn   = blockIdx.x;          // 0 .. B*N-1
    const int lane = threadIdx.x & 31;
    const int wave = threadIdx.x >> 5;    // 0..7
    const int c4   = lane << 2;           // channel base: 0..124

    const v4f cb = *(const v4f*)(combined + bn * HH + c4);
    const v4f g  = *(const v4f*)(gamma + c4);
    const v4f bt = *(const v4f*)(beta + c4);

    for (int t = wave; t < TT; t += 8) {
        const v4f tw = *(const v4f*)(timeW + t * HH + c4);
        v4f h;
        h.x = fmaxf(cb.x + tw.x, 0.0f);
        h.y = fmaxf(cb.y + tw.y, 0.0f);
        h.z = fmaxf(cb.z + tw.z, 0.0f);
        h.w = fmaxf(cb.w + tw.w, 0.0f);

        float s  = h.x + h.y + h.z + h.w;
        float s2 = h.x * h.x + h.y * h.y + h.z * h.z + h.w * h.w;
#pragma unroll
        for (int off = 16; off >= 1; off >>= 1) {
            s  += __shfl_xor(s, off, 32);
            s2 += __shfl_xor(s2, off, 32);
        }
        const float mean = s * (1.0f / (float)HH);
        const float var  = s2 * (1.0f / (float)HH) - mean * mean;
        const float rstd = rsqrtf(var + LN_EPS);

        v4f o;
        o.x = (h.x - mean) * rstd * g.x + bt.x;
        o.y = (h.y - mean) * rstd * g.y + bt.y;
        o.z = (h.z - mean) * rstd * g.z + bt.z;
        o.w = (h.w - mean) * rstd * g.w + bt.w;
        __builtin_nontemporal_store(o, (v4f*)(out + (bn * TT + t) * HH + c4));
    }
}

extern "C" void kernel_launch(void* const* d_in, const int* in_sizes, int n_in,
                              void* d_out, int out_size, void* d_ws, size_t ws_size,
                              hipStream_t stream) {
    (void)in_sizes; (void)n_in; (void)out_size; (void)ws_size;

    const float* x        = (const float*)d_in[0];   // (B, N, T)
    const float* Wp       = (const float*)d_in[1];   // (T, H)
    const float* bp       = (const float*)d_in[2];   // (H,)
    const float* Wf       = (const float*)d_in[3];   // (3H, H)
    const float* bf       = (const float*)d_in[4];   // (H,)
    const float* gamma    = (const float*)d_in[5];   // (H,)
    const float* beta     = (const float*)d_in[6];   // (H,)
    const float* node_emb = (const float*)d_in[7];   // (N, H)
    const float* time_emb = (const float*)d_in[8];   // (T, H)
    float* out = (float*)d_out;                      // (B, N, T, H)

    const float* W1 = Wf;                 // rows [0, H)
    const float* W2 = Wf + HH * HH;       // rows [H, 2H)
    const float* W3 = Wf + 2 * HH * HH;   // rows [2H, 3H)

    // Workspace layout (floats)
    float* proj     = (float*)d_ws;                       // (B*N, H)  = 2 MB
    float* combined = proj + (size_t)BB * NN * HH;        // (B*N, H)  = 2 MB
    float* nodeW    = combined + (size_t)BB * NN * HH;    // (N, H)    = 128 KB
    float* timeW    = nodeW + (size_t)NN * HH;            // (T, H)    = 48 KB

    const dim3 wb(32, 1, 1);

    // nodeW = node_emb @ W2                     (256 x 128, K=128)
    wmma_gemm_f32<false><<<dim3(HH / 16, NN / 16), wb, 0, stream>>>(
        node_emb, W2, nodeW, NN, HH, HH, nullptr, nullptr, 1);

    // timeW = time_emb @ W3                     (96 x 128, K=128)
    wmma_gemm_f32<false><<<dim3(HH / 16, TT / 16), wb, 0, stream>>>(
        time_emb, W3, timeW, TT, HH, HH, nullptr, nullptr, 1);

    // proj = x @ Wp + bp                        (4096 x 128, K=96); x is streamed once -> NT loads
    wmma_gemm_f32<true><<<dim3(HH / 16, (BB * NN) / 16), wb, 0, stream>>>(
        x, Wp, proj, BB * NN, HH, TT, bp, nullptr, 1);

    // combined = proj @ W1 + bf + nodeW[row % 256]   (4096 x 128, K=128)
    wmma_gemm_f32<false><<<dim3(HH / 16, (BB * NN) / 16), wb, 0, stream>>>(
        proj, W1, combined, BB * NN, HH, HH, bf, nodeW, NN);

    // out = LN(relu(combined[bn] + timeW[t])) * gamma + beta
    fused_bias_relu_ln<<<dim3(BB * NN), dim3(256), 0, stream>>>(
        combined, timeW, gamma, beta, out);
}